// CTRMultiEmbedding_60696477827085
// MI455X (gfx1250) — compile-verified
//
#include <hip/hip_runtime.h>

typedef __attribute__((ext_vector_type(2))) float v2f;
typedef __attribute__((ext_vector_type(8))) float v8f;

#define B_    8
#define L_    256
#define D_    64
#define HOURS 168

// -------------------------------------------------------------------------
// joint_embedding: out[n,d] = emb_t[tidx]+emb_l[loc]+emb_u[user]  (gather)
// -------------------------------------------------------------------------
__global__ __launch_bounds__(256) void joint_kernel(
    const int*   __restrict__ traj,   // (B,L,3) int32
    const float* __restrict__ emb_t,  // (169,64)
    const float* __restrict__ emb_l,  // (50000,64)
    const float* __restrict__ emb_u,  // (10000,64)
    float*       __restrict__ out)    // (B*L,64)
{
    const int idx = blockIdx.x * blockDim.x + threadIdx.x;   // 0 .. B*L*D-1
    const int n = idx >> 6;
    const int d = idx & 63;
    const int user = traj[n * 3 + 0];
    const int loc  = traj[n * 3 + 1];
    const int t    = traj[n * 3 + 2];
    // python-style modulo: (t-1) % 168, then +1
    const int tidx = ((t - 1) % HOURS + HOURS) % HOURS + 1;
    out[idx] = emb_t[tidx * D_ + d] + emb_l[loc * D_ + d] + emb_u[user * D_ + d];
}

// -------------------------------------------------------------------------
// delta_embedding via V_WMMA_F32_16X16X4_F32:
//   out[r,:] = (1xK=8 coeff) @ (8x64 embedding matrix), split into two K=4
//   WMMAs (mask==0 rows, then mask==1 rows, accumulating into C).
//   One wave: 16 rows x 64 cols = 4 column tiles x 2 WMMAs.
// -------------------------------------------------------------------------
__global__ __launch_bounds__(256) void delta_kernel(
    const float* __restrict__ mat,     // (B,L,L,2)  -> float2 per row r
    const int*   __restrict__ tlen,    // (B,)
    const float* __restrict__ emb_su,  // (2,64)
    const float* __restrict__ emb_sl,  // (2,64)
    const float* __restrict__ emb_tu,  // (2,64)
    const float* __restrict__ emb_tl,  // (2,64)
    float*       __restrict__ out)     // (B*L*L, 64)
{
    const int lane = threadIdx.x & 31;
    const int wave = blockIdx.x * (blockDim.x >> 5) + (threadIdx.x >> 5);
    const long base = (long)wave * 16;          // first flattened row of tile
    const int  M    = lane & 15;                // row within tile / column idx
    const long r    = base + M;

    // decompose r -> (b, i, j); b and i are wave-uniform within a tile
    const int b = (int)(r >> 16);               // / (L*L)
    const int i = (int)((r >> 8) & 255);
    const int j = (int)(r & 255);
    const int len = tlen[b];
    const float fm = ((i < len) && (j < len)) ? 1.0f : 0.0f;

    // per-row deltas; low half-lanes consume ds (K=0,1), high half dt (K=2,3)
    const float2 dsdt = ((const float2*)mat)[r];
    const bool  lo = (lane < 16);
    const float u  = lo ? dsdt.x : dsdt.y;

    // A operands (16x4 f32): VGPR0 = K even, VGPR1 = K odd
    v2f a0, a1;
    a0.x = (1.0f - fm) * (1.0f - u);            // K=0 / K=2, mask==0 block
    a0.y = (1.0f - fm) * u;                     // K=1 / K=3
    a1.x = fm * (1.0f - u);                     // mask==1 block
    a1.y = fm * u;

    // B operand row sources: K0/K1 = sl/su (low lanes), K2/K3 = tl/tu (high)
    const float* e_x = lo ? emb_sl : emb_tl;
    const float* e_y = lo ? emb_su : emb_tu;

    float* outp = out + base * D_;

#pragma unroll
    for (int t = 0; t < 4; ++t) {
        const int d = t * 16 + M;               // output column
        v2f b0, b1;
        b0.x = e_x[d];        b0.y = e_y[d];        // mask==0 rows
        b1.x = e_x[D_ + d];   b1.y = e_y[D_ + d];   // mask==1 rows

        v8f c = {};
        c = __builtin_amdgcn_wmma_f32_16x16x4_f32(false, a0, false, b0,
                                                  (short)0, c, false, false);
        c = __builtin_amdgcn_wmma_f32_16x16x4_f32(false, a1, false, b1,
                                                  (short)0, c, false, false);

        // C/D layout: VGPR v -> row v (lanes 0-15) or v+8 (lanes 16-31)
        const int rowoff = lo ? 0 : 8;
#pragma unroll
        for (int v = 0; v < 8; ++v) {
            __builtin_nontemporal_store(c[v], &outp[(long)(v + rowoff) * D_ + d]);
        }
    }
}

// -------------------------------------------------------------------------
extern "C" void kernel_launch(void* const* d_in, const int* in_sizes, int n_in,
                              void* d_out, int out_size, void* d_ws, size_t ws_size,
                              hipStream_t stream) {
    const int*   traj    = (const int*)  d_in[0];   // (B,L,3)
    const float* mat     = (const float*)d_in[1];   // (B,L,L,2)
    const int*   tlen    = (const int*)  d_in[2];   // (B,)
    const float* emb_t   = (const float*)d_in[3];
    const float* emb_l   = (const float*)d_in[4];
    const float* emb_u   = (const float*)d_in[5];
    const float* emb_su  = (const float*)d_in[6];
    const float* emb_sl  = (const float*)d_in[7];
    const float* emb_tu  = (const float*)d_in[8];
    const float* emb_tl  = (const float*)d_in[9];

    float* out_joint = (float*)d_out;                    // B*L*D floats
    float* out_delta = out_joint + (B_ * L_ * D_);       // B*L*L*D floats

    // joint: 131072 elements, 1 thread each
    joint_kernel<<<(B_ * L_ * D_) / 256, 256, 0, stream>>>(
        traj, emb_t, emb_l, emb_u, out_joint);

    // delta: 524288 rows / 16 rows-per-wave = 32768 waves; 8 waves/block
    const int waves  = (B_ * L_ * L_) / 16;
    const int blocks = waves / 8;                        // 4096
    delta_kernel<<<blocks, 256, 0, stream>>>(
        mat, tlen, emb_su, emb_sl, emb_tu, emb_tl, out_delta);
}